// DiscretisedDiffusion_21887153340869
// MI455X (gfx1250) — compile-verified
//
#include <hip/hip_runtime.h>
#include <math.h>

#define K_BINS 8192
#define D_IN   16384
#define HIDDEN 2048
#define LEAKY  0.01f
#define TMIN   1e-10f
#define SQRT2F 1.41421356237309504880f
#define WAVES_PER_BLOCK 8

typedef __attribute__((ext_vector_type(2))) float v2f;
typedef __attribute__((ext_vector_type(4))) float v4f;
typedef __attribute__((ext_vector_type(8))) float v8f;

#if __has_builtin(__builtin_amdgcn_wmma_f32_16x16x4_f32)
#define USE_WMMA 1
#endif

// ---------------------------------------------------------------------------
// GEMV via WMMA: y[n_base..n_base+15] = sum_k x[k] * W[k*ld + n] (+ epilogue)
// 8 waves split K; each wave runs two independent V_WMMA_F32_16X16X4_F32
// accumulator chains (K stepped by 8) so the weight loads pipeline.
// A-row-0 = x chunk (other rows zero), B = 4x16 slab of W (non-temporal:
// each weight byte is used exactly once, don't pollute the 192MB L2).
// Row 0 of D lives in C-VGPR0, lanes 0-15. LDS cross-wave reduction.
//   mode 0: y = leaky_relu(sum + bias[n] + t*extra_row[n])   (hidden layer)
//   mode 1: y = sum + bias[n]                                 (output layer)
// ---------------------------------------------------------------------------
__global__ __launch_bounds__(256) void gemv_wmma_kernel(
    const float* __restrict__ x, const float* __restrict__ W,
    const float* __restrict__ bias, const float* __restrict__ t_ptr,
    const float* __restrict__ extra_row, float* __restrict__ y,
    int K_total, int ld, int mode)
{
    const int lane   = threadIdx.x & 31;
    const int wave   = threadIdx.x >> 5;
    const int n_base = blockIdx.x * 16;
    const int col    = n_base + (lane & 15);
    const int kc      = K_total / WAVES_PER_BLOCK;   // divisible by 8
    const int k_begin = wave * kc;
    const int k_end   = k_begin + kc;
    const int r0      = (lane >> 4) * 2;       // K sub-row pair for this half-wave
    const bool head   = (lane & 15) == 0;      // lanes 0 and 16 carry A row 0

    __shared__ float red[WAVES_PER_BLOCK][16];

#ifdef USE_WMMA
    v8f c0 = {0.f, 0.f, 0.f, 0.f, 0.f, 0.f, 0.f, 0.f};
    v8f c1 = {0.f, 0.f, 0.f, 0.f, 0.f, 0.f, 0.f, 0.f};
    for (int k = k_begin; k < k_end; k += 8) {
        const size_t base0 = (size_t)(k + r0) * ld + col;
        const size_t base1 = (size_t)(k + 4 + r0) * ld + col;
        // B (4x16): VGPR0 = rows {0,2}, VGPR1 = rows {1,3} per half-wave.
        // Non-temporal: single-use streaming weights.
        v2f b0, b1;
        b0.x = __builtin_nontemporal_load(W + base0);
        b0.y = __builtin_nontemporal_load(W + base0 + ld);
        b1.x = __builtin_nontemporal_load(W + base1);
        b1.y = __builtin_nontemporal_load(W + base1 + ld);
        // A (16x4): only row M=0 is non-zero (the GEMV input vector)
        v2f a0 = {0.f, 0.f};
        v2f a1 = {0.f, 0.f};
        if (head) {
            a0.x = x[k + r0];     a0.y = x[k + r0 + 1];
            a1.x = x[k + 4 + r0]; a1.y = x[k + 4 + r0 + 1];
        }
        c0 = __builtin_amdgcn_wmma_f32_16x16x4_f32(
                false, a0, false, b0, (short)0, c0, false, false);
        c1 = __builtin_amdgcn_wmma_f32_16x16x4_f32(
                false, a1, false, b1, (short)0, c1, false, false);
    }
    if (lane < 16) red[wave][lane] = c0[0] + c1[0];  // D row 0 = VGPR0, lanes 0-15
#else
    // Fallback (no CDNA5 wmma builtin): scalar FMA, 16 active lanes per wave.
    float acc = 0.f;
    if (lane < 16) {
        for (int k = k_begin; k < k_end; ++k)
            acc += x[k] * W[(size_t)k * ld + col];
    }
    if (lane < 16) red[wave][lane] = acc;
#endif
    __syncthreads();

    if (wave == 0 && lane < 16) {
        float s = 0.f;
        #pragma unroll
        for (int wv = 0; wv < WAVES_PER_BLOCK; ++wv) s += red[wv][lane];
        const int n = n_base + lane;
        s += bias[n];
        if (mode == 0) {
            s += t_ptr[0] * extra_row[n];          // the "+1" row times t
            y[n] = (s > 0.f) ? s : LEAKY * s;      // leaky_relu
        } else {
            y[n] = s;
        }
    }
}

// ---------------------------------------------------------------------------
// Per-bin parameters: mu_x[i], 1/(sigma_x[i]*sqrt2)
// ---------------------------------------------------------------------------
__global__ void prep_kernel(const float* __restrict__ mu,
                            const float* __restrict__ t,
                            const float* __restrict__ gamma,
                            const float* __restrict__ outv,
                            float* __restrict__ mux,
                            float* __restrict__ invd)
{
    const int i = blockIdx.x * blockDim.x + threadIdx.x;
    if (i >= K_BINS) return;
    const float g     = gamma[0];
    const float tt    = t[0];
    const float p_eps = 1.f / (1.f - g);
    const float p_mu  = g - p_eps;
    const float mu_eps = outv[i];
    const float ln_sig = outv[K_BINS + i];
    float mu_x  = powf(mu[i], p_mu) * powf(mu_eps, p_eps);
    float sigma = powf(1.f - g, -0.5f) * expf(0.5f * ln_sig);
    const bool use_nn = (tt >= TMIN);
    mu_x  = use_nn ? mu_x  : 0.f;
    sigma = use_nn ? sigma : 1.f;
    mux[i]  = mu_x;
    invd[i] = 1.f / (sigma * SQRT2F);
}

// ---------------------------------------------------------------------------
// out[i,j] = F(kr[j]) - F(kl[j]),  F(x) = x>=1 ? 1 : 0.5*(1+erf((x-mu)/(sig*sqrt2)))
// kr[j] == kl[j+1]  => running difference: 9 erfs per 8 outputs.
// If kl[j0] >= 1 for the whole 8-wide strip, output is exactly 0 (1 - 1).
// 256 MB of single-use output: non-temporal streaming stores.
// ---------------------------------------------------------------------------
__global__ __launch_bounds__(256) void cdf_kernel(
    const float* __restrict__ mux, const float* __restrict__ invd,
    float* __restrict__ out)
{
    const int i  = blockIdx.y;
    const int j0 = (blockIdx.x * 256 + threadIdx.x) * 8;
    v4f* dst = (v4f*)(out + (size_t)i * K_BINS + j0);

    const float scale = 2.0f / (float)(K_BINS - 1);
    const float x_lo  = (float)(j0 - 1) * scale;   // kl[j0]
    if (x_lo >= 1.0f) {                            // both CDFs clamp to 1
        v4f z = {0.f, 0.f, 0.f, 0.f};
        __builtin_nontemporal_store(z, dst);
        __builtin_nontemporal_store(z, dst + 1);
        return;
    }

    const float m   = mux[i];
    const float inv = invd[i];

    float Fprev = 0.5f * (1.f + erff((x_lo - m) * inv));
    float r[8];
    #pragma unroll
    for (int q = 0; q < 8; ++q) {
        const float xx = (float)(j0 + q) * scale;  // kr[j0+q]
        const float Fc = (xx >= 1.f) ? 1.f
                       : 0.5f * (1.f + erff((xx - m) * inv));
        r[q]  = Fc - Fprev;
        Fprev = Fc;
    }
    v4f lo = {r[0], r[1], r[2], r[3]};
    v4f hi = {r[4], r[5], r[6], r[7]};
    __builtin_nontemporal_store(lo, dst);
    __builtin_nontemporal_store(hi, dst + 1);
}

// ---------------------------------------------------------------------------
extern "C" void kernel_launch(void* const* d_in, const int* in_sizes, int n_in,
                              void* d_out, int out_size, void* d_ws, size_t ws_size,
                              hipStream_t stream)
{
    const float* mu    = (const float*)d_in[0];
    const float* t     = (const float*)d_in[1];
    const float* gamma = (const float*)d_in[2];
    const float* W1    = (const float*)d_in[3];
    const float* b1    = (const float*)d_in[4];
    const float* W2    = (const float*)d_in[5];
    const float* b2    = (const float*)d_in[6];
    float* out = (float*)d_out;

    float* ws   = (float*)d_ws;
    float* h    = ws;                       // HIDDEN
    float* outv = h    + HIDDEN;            // 2*K_BINS
    float* mux  = outv + 2 * K_BINS;        // K_BINS
    float* invd = mux  + K_BINS;            // K_BINS

    // h = leaky_relu([mu; t] @ W1 + b1)
    gemv_wmma_kernel<<<HIDDEN / 16, 256, 0, stream>>>(
        mu, W1, b1, t, W1 + (size_t)D_IN * HIDDEN, h, D_IN, HIDDEN, 0);
    // outv = h @ W2 + b2
    gemv_wmma_kernel<<<D_IN / 16, 256, 0, stream>>>(
        h, W2, b2, nullptr, nullptr, outv, HIDDEN, D_IN, 1);
    // per-bin mu_x, 1/(sigma*sqrt2)
    prep_kernel<<<K_BINS / 256, 256, 0, stream>>>(mu, t, gamma, outv, mux, invd);
    // 8192x8192 CDF-difference matrix
    cdf_kernel<<<dim3(K_BINS / 2048, K_BINS), 256, 0, stream>>>(mux, invd, out);
}